// TAlayer_9251359556269
// MI455X (gfx1250) — compile-verified
//
#include <hip/hip_runtime.h>
#include <stdint.h>

// Tropical (max-plus) matmul: out[n,m] = max_k(|x[n,k]| + |w[m,k]|)
// N=M=4096, K=512, fp32. VALU-bound; data movement via CDNA5 Tensor Data Mover
// (tensor_load_to_lds, TENSORcnt) with double-buffered 128x128x32 tiles.

#define N_DIM 4096
#define M_DIM 4096
#define K_DIM 512
#define BM 128
#define BN 128
#define BK 32
#define WPITCH 36                      // padded w-tile row pitch (dwords): conflict-free b128
#define XTILE_DW (BM * BK)             // 4096 dwords (unpadded: broadcast access)
#define WTILE_DW (BN * WPITCH)         // 4608 dwords
#define BUF_DW   (XTILE_DW + WTILE_DW) // 8704 dwords per double-buffer slot
#define KTILES   (K_DIM / BK)          // 16

typedef unsigned int v4u __attribute__((ext_vector_type(4)));
typedef int          v8i __attribute__((ext_vector_type(8)));
typedef int          v4i __attribute__((ext_vector_type(4)));

// Build a 2D TDM descriptor (D#) per CDNA5 ISA ch.8 and issue TENSOR_LOAD_TO_LDS.
// Tile: tile_d0 (contiguous dwords) x tile_d1 rows from a row-major tensor of
// width tens_d0 / height tens_d1 dwords, row stride stride0 dwords.
__device__ __forceinline__ void tdm_load_2d(const float* gsrc, uint32_t lds_byte,
                                            uint32_t tile_d0, uint32_t tile_d1,
                                            uint32_t tens_d0, uint32_t tens_d1,
                                            uint32_t stride0, uint32_t pad_bits)
{
    uint64_t ga = (uint64_t)(uintptr_t)gsrc;
    v4u g0;
    g0.x = 1u;                                                  // count=1 (valid user D#)
    g0.y = lds_byte;                                            // lds_addr (bytes)
    g0.z = (uint32_t)ga;                                        // global_addr[31:0]
    g0.w = (uint32_t)((ga >> 32) & 0x01FFFFFFu) | (2u << 30);   // addr[56:32] | type=2
    v8i g1;
    g1[0] = (int)((2u << 16) | pad_bits);                       // data_size=4B | pad ctl
    g1[1] = (int)(tens_d0 << 16);                               // tensor_dim0[15:0] @ bits63:48
    g1[2] = (int)((tens_d0 >> 16) | (tens_d1 << 16));           // dim0[31:16] | dim1[15:0]
    g1[3] = (int)((tens_d1 >> 16) | (tile_d0 << 16));           // dim1[31:16] | tile_dim0
    g1[4] = (int)(tile_d1);                                     // tile_dim1 | tile_dim2=0
    g1[5] = (int)(stride0);                                     // tensor_dim0_stride[31:0]
    g1[6] = 0;                                                  // stride0 hi | dim1_stride lo
    g1[7] = 0;
    v4i g2 = (v4i)0;                                            // group2: unused (2D tile)
    v4i g3 = (v4i)0;                                            // group3: unused
    v8i g4 = (v8i)0;                                            // extra group (clang-23 6-arg form)
    __builtin_amdgcn_tensor_load_to_lds(g0, g1, g2, g3, g4, 0);
}

__global__ __launch_bounds__(256)
void tropical_mm_kernel(const float* __restrict__ X, const float* __restrict__ W,
                        float* __restrict__ Out)
{
    __shared__ float smem[2 * BUF_DW];   // 69632 bytes: 2 buffers x (x-tile + padded w-tile)

    const int tid = threadIdx.x;
    const int tx  = tid & 15;            // m-side lane: columns m = tx + 16*j
    const int ty  = tid >> 4;            // n-side lane: rows    n = ty + 16*i

    const float* Xblk = X + (size_t)blockIdx.y * BM * K_DIM;
    const float* Wblk = W + (size_t)blockIdx.x * BN * K_DIM;

    const uint32_t lds_base = (uint32_t)(uintptr_t)(&smem[0]);
    // w-tile padding: pad_enable | interval code 4 (=32 dwords) | amount code 3 (=4 dwords)
    const uint32_t padw = (1u << 20) | (4u << 22) | (3u << 25);

    float acc[8][8];
#pragma unroll
    for (int i = 0; i < 8; ++i)
#pragma unroll
        for (int j = 0; j < 8; ++j) acc[i][j] = -__builtin_inff();

    // Prologue: wave 0 issues TDM loads for K-tile 0 into buffer 0 (TENSORcnt = 2).
    if (tid < 32) {
        tdm_load_2d(Xblk, lds_base,                 BK, BM, K_DIM, N_DIM, K_DIM, 0);
        tdm_load_2d(Wblk, lds_base + XTILE_DW * 4u, BK, BN, K_DIM, M_DIM, K_DIM, padw);
    }

    for (int t = 0; t < KTILES; ++t) {
        const int b = t & 1;
        if (tid < 32) {
            if (t + 1 < KTILES) {
                const int nb = b ^ 1;
                const uint32_t base = lds_base + (uint32_t)(nb * BUF_DW) * 4u;
                tdm_load_2d(Xblk + (t + 1) * BK, base,                 BK, BM, K_DIM, N_DIM, K_DIM, 0);
                tdm_load_2d(Wblk + (t + 1) * BK, base + XTILE_DW * 4u, BK, BN, K_DIM, M_DIM, K_DIM, padw);
                // In-order TENSORcnt: <=2 outstanding means tile t's two loads are done.
                __builtin_amdgcn_s_wait_tensorcnt(2);
            } else {
                __builtin_amdgcn_s_wait_tensorcnt(0);
            }
        }
        __syncthreads();                       // publish TDM-written LDS to all waves
        __asm__ __volatile__("" ::: "memory"); // compiler fence: LDS written outside its view

        const float* xs = smem + b * BUF_DW;
        const float* ws = xs + XTILE_DW;

        for (int kq = 0; kq < BK; kq += 4) {
            float4 av[8], bv[8];
#pragma unroll
            for (int i = 0; i < 8; ++i)        // 2-address broadcast across the wave
                av[i] = *(const float4*)(xs + (ty + 16 * i) * BK + kq);
#pragma unroll
            for (int j = 0; j < 8; ++j)        // pitch 36 + stride-16 mapping: conflict-free
                bv[j] = *(const float4*)(ws + (tx + 16 * j) * WPITCH + kq);
#pragma unroll
            for (int i = 0; i < 8; ++i)
#pragma unroll
                for (int j = 0; j < 8; ++j) {  // abs folds into VOP3 |src| modifiers
                    acc[i][j] = fmaxf(acc[i][j], fabsf(av[i].x) + fabsf(bv[j].x));
                    acc[i][j] = fmaxf(acc[i][j], fabsf(av[i].y) + fabsf(bv[j].y));
                    acc[i][j] = fmaxf(acc[i][j], fabsf(av[i].z) + fabsf(bv[j].z));
                    acc[i][j] = fmaxf(acc[i][j], fabsf(av[i].w) + fabsf(bv[j].w));
                }
        }
        __syncthreads();                       // all reads of buffer b done before reuse
    }

    // Epilogue: out[n, m]; 16 consecutive lanes cover consecutive m -> coalesced stores.
    float* outp = Out + (size_t)blockIdx.y * BM * M_DIM + (size_t)blockIdx.x * BN;
#pragma unroll
    for (int i = 0; i < 8; ++i) {
        const int n = ty + 16 * i;
#pragma unroll
        for (int j = 0; j < 8; ++j)
            outp[(size_t)n * M_DIM + tx + 16 * j] = acc[i][j];
    }
}

extern "C" void kernel_launch(void* const* d_in, const int* in_sizes, int n_in,
                              void* d_out, int out_size, void* d_ws, size_t ws_size,
                              hipStream_t stream) {
    (void)in_sizes; (void)n_in; (void)out_size; (void)d_ws; (void)ws_size;
    const float* x = (const float*)d_in[0];   // (N, K) fp32
    const float* w = (const float*)d_in[1];   // (M, K) fp32
    float* out = (float*)d_out;               // (N, M) fp32
    dim3 grid(M_DIM / BN, N_DIM / BM);        // 32 x 32 blocks
    tropical_mm_kernel<<<grid, 256, 0, stream>>>(x, w, out);
}